// LiquidNet_10720238371159
// MI455X (gfx1250) — compile-verified
//
#include <hip/hip_runtime.h>
#include <stdint.h>

#define BB   1024
#define II   128
#define NN   256
#define UNF  6
#define ROWS 8          // batch rows per block
#define CT   8          // pre-neuron rows per LDS chunk
#define L2E  1.4426950408889634f

typedef float v2f __attribute__((ext_vector_type(2)));
typedef float v8f __attribute__((ext_vector_type(8)));

// ---- CDNA5 async global->LDS copy (ASYNCcnt path) ----
__device__ __forceinline__ void async_b128(uint32_t lds_off, uint64_t gaddr) {
    asm volatile("global_load_async_to_lds_b128 %0, %1, off"
                 :: "v"(lds_off), "v"(gaddr) : "memory");
}
__device__ __forceinline__ void wait_async_le8() {
    asm volatile("s_wait_asynccnt 8" ::: "memory");
}
__device__ __forceinline__ void wait_async_le0() {
    asm volatile("s_wait_asynccnt 0" ::: "memory");
}

// Copy CT rows x 256 cols (f32) of 4 matrices into one LDS buffer.
// CT*256*4B = 8KB per matrix = 512 b128 transfers = 2 per thread (256 thr).
__device__ __forceinline__ void issue_chunk(const float* p0, const float* p1,
                                            const float* p2, const float* p3,
                                            int row0, float (*buf)[CT * NN], int tid) {
    const float* src[4] = { p0 + row0 * NN, p1 + row0 * NN,
                            p2 + row0 * NN, p3 + row0 * NN };
#pragma unroll
    for (int m = 0; m < 4; ++m) {
        uint64_t g = (uint64_t)(uintptr_t)src[m];
        uint32_t l = (uint32_t)(uintptr_t)&buf[m][0];
#pragma unroll
        for (int t = 0; t < 2; ++t) {
            uint32_t off = (uint32_t)(tid + t * 256) * 16u;
            async_b128(l + off, g + off);
        }
    }
}

__global__ void __launch_bounds__(256)
ltc_fused_kernel(const float* __restrict__ inputs, const float* __restrict__ state,
                 const float* __restrict__ map_w,  const float* __restrict__ map_b,
                 const float* __restrict__ smu,    const float* __restrict__ ssg,
                 const float* __restrict__ sW,     const float* __restrict__ ser,
                 const float* __restrict__ rmu,    const float* __restrict__ rsg,
                 const float* __restrict__ rW,     const float* __restrict__ rer,
                 const float* __restrict__ vleak,  const float* __restrict__ gleak,
                 const float* __restrict__ cm_t,   float* __restrict__ out)
{
    // transposed layouts: row = 8 contiguous floats (32B) -> b128 broadcast reads
    __shared__ float sx[II][ROWS];            // mapped inputs x^T (4 KB)
    __shared__ float sv[NN][ROWS];            // state v^T         (8 KB)
    __shared__ float sbuf[2][4][CT * NN];     // param dbl-buffer  (64 KB)

    const int tid = threadIdx.x;
    const int n   = tid;                      // post-neuron column owned by thread
    const int r0  = blockIdx.x * ROWS;

    // ---- Stage 0: x = inputs * diag(map_w) + map_b via V_WMMA_F32_16X16X4_F32
    // (exact in f32: only the diagonal K==N term contributes per 16-col tile)
    {
        const int wave = tid >> 5, lane = tid & 31;
        const int t    = wave * 16;               // 8 waves cover I=128 cols
        const int m    = lane & 15;               // A row (batch)
        const int half = lane >> 4;
        const int kb   = half * 2;                // K lanes-split
        const int nn2  = lane & 15;               // B/C column
        int row = r0 + m; if (row > BB - 1) row = BB - 1;
        const float w = map_w[t + nn2];
        v8f c = {0.f, 0.f, 0.f, 0.f, 0.f, 0.f, 0.f, 0.f};
#pragma unroll
        for (int s = 0; s < 4; ++s) {             // K = 16 in steps of 4
            const int k0 = 4 * s + kb;
            v2f a, b;
            a.x = inputs[row * II + t + k0];
            a.y = inputs[row * II + t + k0 + 1];
            b.x = (k0     == nn2) ? w : 0.f;      // diagonal B tile
            b.y = (k0 + 1 == nn2) ? w : 0.f;
            c = __builtin_amdgcn_wmma_f32_16x16x4_f32(false, a, false, b,
                                                      (short)0, c, false, false);
        }
        if (half == 0) {                          // lanes 0-15 hold M = r (0..7)
            const float bb = map_b[t + nn2];
            float4* p = (float4*)&sx[t + nn2][0];
            p[0] = make_float4(c[0] + bb, c[1] + bb, c[2] + bb, c[3] + bb);
            p[1] = make_float4(c[4] + bb, c[5] + bb, c[6] + bb, c[7] + bb);
        }
    }

    // per-column constants
    const float cm = cm_t[n];
    const float gl = gleak[n];
    const float gv = gl * vleak[n];

    float sn[ROWS], sd[ROWS];                     // sensory num/den accumulators
#pragma unroll
    for (int b = 0; b < ROWS; ++b) { sn[b] = 0.f; sd[b] = 0.f; }

    float vcur[ROWS];
#pragma unroll
    for (int b = 0; b < ROWS; ++b) vcur[b] = state[(r0 + b) * NN + n];
    {
        float4* pv = (float4*)&sv[n][0];
        pv[0] = make_float4(vcur[0], vcur[1], vcur[2], vcur[3]);
        pv[1] = make_float4(vcur[4], vcur[5], vcur[6], vcur[7]);
    }

    // ---- Stage 1: sensory synapses (I=128 pre-rows, async double-buffered)
    {
        const int C = II / CT;                    // 16 chunks
        issue_chunk(smu, ssg, sW, ser, 0, sbuf[0], tid);
        for (int c = 0; c < C; ++c) {
            if (c + 1 < C) {
                issue_chunk(smu, ssg, sW, ser, (c + 1) * CT, sbuf[(c + 1) & 1], tid);
                wait_async_le8();                 // oldest 8 (chunk c) complete
            } else {
                wait_async_le0();
            }
            __syncthreads();                      // chunk c visible to all waves
            const float* bmu = &sbuf[c & 1][0][0];
            const float* bsg = &sbuf[c & 1][1][0];
            const float* bW  = &sbuf[c & 1][2][0];
            const float* ber = &sbuf[c & 1][3][0];
#pragma unroll
            for (int jj = 0; jj < CT; ++jj) {
                const float mu  = bmu[jj * NN + n];
                const float s2  = bsg[jj * NN + n] * L2E;
                const float W   = bW [jj * NN + n];
                const float we  = W * ber[jj * NN + n];
                const float pp  = s2 * mu;        // arg = fma(-s2, x, s2*mu)
                const float ns2 = -s2;
                const int   i   = c * CT + jj;
                const float4 xa = *(const float4*)&sx[i][0];   // b128 broadcast
                const float4 xb = *(const float4*)&sx[i][4];
                const float xv[ROWS] = { xa.x, xa.y, xa.z, xa.w,
                                         xb.x, xb.y, xb.z, xb.w };
#pragma unroll
                for (int b = 0; b < ROWS; ++b) {
                    const float e  = __builtin_amdgcn_exp2f(__builtin_fmaf(ns2, xv[b], pp));
                    const float sg = __builtin_amdgcn_rcpf(1.f + e);   // sigmoid
                    sn[b] += we * sg;
                    sd[b] += W  * sg;
                }
            }
            __syncthreads();                      // all readers done with buf (c&1)
        }
    }

    // ---- Stage 2: six fused unfolds (N=256 pre-rows each, async double-buffered)
    const int C = NN / CT;                        // 32 chunks
    for (int u = 0; u < UNF; ++u) {
        float num[ROWS], den[ROWS];
#pragma unroll
        for (int b = 0; b < ROWS; ++b) { num[b] = sn[b]; den[b] = sd[b]; }

        issue_chunk(rmu, rsg, rW, rer, 0, sbuf[0], tid);
        for (int c = 0; c < C; ++c) {
            if (c + 1 < C) {
                issue_chunk(rmu, rsg, rW, rer, (c + 1) * CT, sbuf[(c + 1) & 1], tid);
                wait_async_le8();
            } else {
                wait_async_le0();
            }
            __syncthreads();
            const float* bmu = &sbuf[c & 1][0][0];
            const float* bsg = &sbuf[c & 1][1][0];
            const float* bW  = &sbuf[c & 1][2][0];
            const float* ber = &sbuf[c & 1][3][0];
#pragma unroll
            for (int jj = 0; jj < CT; ++jj) {
                const float mu  = bmu[jj * NN + n];
                const float s2  = bsg[jj * NN + n] * L2E;
                const float W   = bW [jj * NN + n];
                const float we  = W * ber[jj * NN + n];
                const float pp  = s2 * mu;
                const float ns2 = -s2;
                const int   j   = c * CT + jj;
                const float4 va = *(const float4*)&sv[j][0];   // b128 broadcast
                const float4 vb = *(const float4*)&sv[j][4];
                const float vvv[ROWS] = { va.x, va.y, va.z, va.w,
                                          vb.x, vb.y, vb.z, vb.w };
#pragma unroll
                for (int b = 0; b < ROWS; ++b) {
                    const float e  = __builtin_amdgcn_exp2f(__builtin_fmaf(ns2, vvv[b], pp));
                    const float sg = __builtin_amdgcn_rcpf(1.f + e);
                    num[b] += we * sg;
                    den[b] += W  * sg;
                }
            }
            __syncthreads();
        }

        // v_new = (cm*v + gleak*vleak + num) / (cm + gleak + den)
#pragma unroll
        for (int b = 0; b < ROWS; ++b) {
            const float nume = cm * vcur[b] + gv + num[b];
            const float deno = cm + gl + den[b];
            vcur[b] = nume / deno;
        }
        // all reads of sv for this unfold finished before the last trailing barrier
        {
            float4* pv = (float4*)&sv[n][0];
            pv[0] = make_float4(vcur[0], vcur[1], vcur[2], vcur[3]);
            pv[1] = make_float4(vcur[4], vcur[5], vcur[6], vcur[7]);
        }
        __syncthreads();
    }

    // ---- Output: reference returns (v, v) -> two flat copies
#pragma unroll
    for (int b = 0; b < ROWS; ++b) {
        const int row = r0 + b;
        out[row * NN + n]           = vcur[b];
        out[BB * NN + row * NN + n] = vcur[b];
    }
}

extern "C" void kernel_launch(void* const* d_in, const int* in_sizes, int n_in,
                              void* d_out, int out_size, void* d_ws, size_t ws_size,
                              hipStream_t stream) {
    (void)in_sizes; (void)n_in; (void)d_ws; (void)ws_size; (void)out_size;
    const float* inputs = (const float*)d_in[0];
    const float* state  = (const float*)d_in[1];
    const float* map_w  = (const float*)d_in[2];
    const float* map_b  = (const float*)d_in[3];
    const float* smu    = (const float*)d_in[4];
    const float* ssg    = (const float*)d_in[5];
    const float* sW     = (const float*)d_in[6];
    const float* ser    = (const float*)d_in[7];
    const float* rmu    = (const float*)d_in[8];
    const float* rsg    = (const float*)d_in[9];
    const float* rW     = (const float*)d_in[10];
    const float* rer    = (const float*)d_in[11];
    const float* vleak  = (const float*)d_in[12];
    const float* gleak  = (const float*)d_in[13];
    const float* cm_t   = (const float*)d_in[14];
    float* out = (float*)d_out;

    dim3 grid(BB / ROWS);   // 128 blocks
    dim3 block(256);        // 8 waves (wave32) per block
    ltc_fused_kernel<<<grid, block, 0, stream>>>(inputs, state, map_w, map_b,
                                                 smu, ssg, sW, ser,
                                                 rmu, rsg, rW, rer,
                                                 vleak, gleak, cm_t, out);
}